// Voxelization_47571057771226
// MI455X (gfx1250) — compile-verified
//
#include <hip/hip_runtime.h>
#include <stdint.h>

#define BB  16
#define CC  64
#define NN  65536
#define RES 32
#define NV  32768   // 32^3 voxels

// ---------------------------------------------------------------------------
// CDNA5 async global->LDS helpers (gfx1250). Each lane copies 16 bytes from global address g to LDS address lds.
// Tracked with ASYNCcnt; per-wave s_wait_asynccnt gates consumption.
// ---------------------------------------------------------------------------
__device__ __forceinline__ uint32_t lds_addr_of(const void* p) {
  // Low 32 bits of a generic/shared pointer are the LDS byte offset on gfx1250.
  return (uint32_t)(uintptr_t)p;
}

__device__ __forceinline__ void async_ld_b128(uint32_t lds, const void* g) {
  asm volatile("global_load_async_to_lds_b128 %0, %1, off"
               :
               : "v"(lds), "v"((uint64_t)(uintptr_t)g)
               : "memory");
}

// ---------------------------------------------------------------------------
// Kernel 4 (defined first so it leads the disassembly): scatter-mean.
// One workgroup owns one (batch, channel) pair and a full 128 KB LDS voxel
// slice. Feature/index chunks are double-buffered into per-wave LDS staging
// with async b128 copies (ASYNCcnt pipeline), accumulated with ds_add_f32,
// then divided by counts and stored once, coalesced, with b128 accesses.
// LDS: 128 KB acc + 16 KB staging = 144 KB (fits CDNA5 WGP LDS).
// ---------------------------------------------------------------------------
__global__ void __launch_bounds__(256) k_scatter(const float* __restrict__ features,
                                                 const int* __restrict__ idx,
                                                 const float* __restrict__ cnt,
                                                 float* __restrict__ out) {
  __shared__ __align__(16) float acc[NV];          // 128 KB accumulation slice
  __shared__ __align__(16) float fstage[2][1024];  // 2 x 4 KB feature staging
  __shared__ __align__(16) int   istage[2][1024];  // 2 x 4 KB index staging

  const int c   = blockIdx.x;
  const int b   = blockIdx.y;
  const int tid = threadIdx.x;
  const int e   = tid * 4;             // 4 contiguous elements per thread (b128)

  const float* fb = features + ((size_t)b * CC + c) * NN;
  const int*   ib = idx + (size_t)b * NN;

  // Prologue: chunks 0 and 1 in flight while we zero the accumulator.
  async_ld_b128(lds_addr_of(&fstage[0][e]), fb + 0 * 1024 + e);
  async_ld_b128(lds_addr_of(&istage[0][e]), ib + 0 * 1024 + e);
  async_ld_b128(lds_addr_of(&fstage[1][e]), fb + 1 * 1024 + e);
  async_ld_b128(lds_addr_of(&istage[1][e]), ib + 1 * 1024 + e);

  // Zero the accumulator with ds_store_b128.
  float4* acc4 = (float4*)acc;
  for (int v = tid; v < NV / 4; v += 256)
    acc4[v] = make_float4(0.0f, 0.0f, 0.0f, 0.0f);
  __syncthreads();

  // 64 chunks of 1024 points; each wave consumes only its own staged lanes,
  // so per-wave s_wait_asynccnt is a sufficient guard (no barriers needed).
  for (int k = 0; k < 64; ++k) {
    const int p = k & 1;
    if (k < 63) {
      asm volatile("s_wait_asynccnt 2" ::: "memory");  // chunk k arrived
    } else {
      asm volatile("s_wait_asynccnt 0" ::: "memory");  // drain tail
    }
#pragma unroll
    for (int j = 0; j < 4; ++j) {
      const float val = fstage[p][e + j];
      const int   id  = istage[p][e + j];
      atomicAdd(&acc[id], val);                        // ds_add_f32
    }
    if (k + 2 < 64) {                                  // refill buffer p
      async_ld_b128(lds_addr_of(&fstage[p][e]), fb + (size_t)(k + 2) * 1024 + e);
      async_ld_b128(lds_addr_of(&istage[p][e]), ib + (size_t)(k + 2) * 1024 + e);
    }
  }
  __syncthreads();

  // Fused mean-divide + single coalesced b128 store of the channel slice.
  const float4* cb4 = (const float4*)(cnt + (size_t)b * NV);
  float4*       ob4 = (float4*)(out + ((size_t)b * CC + c) * (size_t)NV);
  for (int v = tid; v < NV / 4; v += 256) {
    const float4 a = acc4[v];
    const float4 q = cb4[v];
    float4 r;
    r.x = a.x / fmaxf(q.x, 1.0f);
    r.y = a.y / fmaxf(q.y, 1.0f);
    r.z = a.z / fmaxf(q.z, 1.0f);
    r.w = a.w / fmaxf(q.w, 1.0f);
    ob4[v] = r;
  }
}

// ---------------------------------------------------------------------------
// Kernel 1: per-batch coordinate means (f64 accumulation for stability)
// ---------------------------------------------------------------------------
__global__ void __launch_bounds__(256) k_mean(const float* __restrict__ coords,
                                              float* __restrict__ mean) {
  const int b = blockIdx.x;
  const float* cb = coords + (size_t)b * 3 * NN;
  double s0 = 0.0, s1 = 0.0, s2 = 0.0;
  for (int n = threadIdx.x; n < NN; n += 256) {
    s0 += cb[n];
    s1 += cb[NN + n];
    s2 += cb[2 * NN + n];
  }
  __shared__ double r0[256], r1[256], r2[256];
  r0[threadIdx.x] = s0; r1[threadIdx.x] = s1; r2[threadIdx.x] = s2;
  __syncthreads();
  for (int s = 128; s > 0; s >>= 1) {
    if (threadIdx.x < s) {
      r0[threadIdx.x] += r0[threadIdx.x + s];
      r1[threadIdx.x] += r1[threadIdx.x + s];
      r2[threadIdx.x] += r2[threadIdx.x + s];
    }
    __syncthreads();
  }
  if (threadIdx.x == 0) {
    mean[b * 3 + 0] = (float)(r0[0] * (1.0 / NN));
    mean[b * 3 + 1] = (float)(r1[0] * (1.0 / NN));
    mean[b * 3 + 2] = (float)(r2[0] * (1.0 / NN));
  }
}

// ---------------------------------------------------------------------------
// Kernel 2: per-batch 2*radius = 2*sqrt(max squared norm) (max is exact)
// ---------------------------------------------------------------------------
__global__ void __launch_bounds__(256) k_radius(const float* __restrict__ coords,
                                                const float* __restrict__ mean,
                                                float* __restrict__ rad2) {
  const int b = blockIdx.x;
  const float* cb = coords + (size_t)b * 3 * NN;
  const float m0 = mean[b * 3 + 0], m1 = mean[b * 3 + 1], m2 = mean[b * 3 + 2];
  float mx = 0.0f;
  for (int n = threadIdx.x; n < NN; n += 256) {
    const float dx = cb[n] - m0;
    const float dy = cb[NN + n] - m1;
    const float dz = cb[2 * NN + n] - m2;
    mx = fmaxf(mx, dx * dx + dy * dy + dz * dz);
  }
  __shared__ float red[256];
  red[threadIdx.x] = mx;
  __syncthreads();
  for (int s = 128; s > 0; s >>= 1) {
    if (threadIdx.x < s)
      red[threadIdx.x] = fmaxf(red[threadIdx.x], red[threadIdx.x + s]);
    __syncthreads();
  }
  if (threadIdx.x == 0) rad2[b] = 2.0f * sqrtf(red[0]);
}

// ---------------------------------------------------------------------------
// Kernel 3: per point -> normalized coords (output 2), voxel idx (ws),
//           float counts via global atomics (exact integer-valued adds)
// ---------------------------------------------------------------------------
__global__ void __launch_bounds__(256) k_points(const float* __restrict__ coords,
                                                const float* __restrict__ mean,
                                                const float* __restrict__ rad2,
                                                float* __restrict__ norm_out,
                                                int* __restrict__ idx_out,
                                                float* __restrict__ cnt) {
  const int t = blockIdx.x * 256 + threadIdx.x;   // 0 .. B*N-1
  const int b = t >> 16;                          // N == 65536
  const int n = t & (NN - 1);
  const float* cb = coords + (size_t)b * 3 * NN;
  const float r2 = rad2[b];
  const float m0 = mean[b * 3 + 0], m1 = mean[b * 3 + 1], m2 = mean[b * 3 + 2];

  float x = (cb[n]          - m0) / r2 + 0.5f;
  float y = (cb[NN + n]     - m1) / r2 + 0.5f;
  float z = (cb[2 * NN + n] - m2) / r2 + 0.5f;
  x = fminf(fmaxf(x * (float)RES, 0.0f), (float)(RES - 1));
  y = fminf(fmaxf(y * (float)RES, 0.0f), (float)(RES - 1));
  z = fminf(fmaxf(z * (float)RES, 0.0f), (float)(RES - 1));

  norm_out[(size_t)(b * 3 + 0) * NN + n] = x;
  norm_out[(size_t)(b * 3 + 1) * NN + n] = y;
  norm_out[(size_t)(b * 3 + 2) * NN + n] = z;

  const int vx = (int)rintf(x);   // v_rndne: round-half-even, matches jnp.round
  const int vy = (int)rintf(y);
  const int vz = (int)rintf(z);
  const int id = (vx * RES + vy) * RES + vz;
  idx_out[t] = id;
  atomicAdd(&cnt[(size_t)b * NV + id], 1.0f);
}

// ---------------------------------------------------------------------------
// Host launcher
// ---------------------------------------------------------------------------
extern "C" void kernel_launch(void* const* d_in, const int* in_sizes, int n_in,
                              void* d_out, int out_size, void* d_ws, size_t ws_size,
                              hipStream_t stream) {
  (void)in_sizes; (void)n_in; (void)out_size; (void)ws_size;

  const float* features = (const float*)d_in[0];   // [B, C, N]
  const float* coords   = (const float*)d_in[1];   // [B, 3, N]

  float* out_vox  = (float*)d_out;                                  // [B, C, 32^3]
  float* out_norm = (float*)d_out + (size_t)BB * CC * NV;           // [B, 3, N]

  float* w_mean = (float*)d_ws;                     // 48 floats
  float* w_rad  = w_mean + 48;                      // 16 floats
  float* w_cnt  = w_mean + 64;                      // B*NV floats (2 MB)
  int*   w_idx  = (int*)(w_cnt + (size_t)BB * NV);  // B*N ints  (4 MB)

  hipMemsetAsync(w_cnt, 0, (size_t)BB * NV * sizeof(float), stream);

  k_mean  <<<BB, 256, 0, stream>>>(coords, w_mean);
  k_radius<<<BB, 256, 0, stream>>>(coords, w_mean, w_rad);
  k_points<<<(BB * NN) / 256, 256, 0, stream>>>(coords, w_mean, w_rad,
                                                out_norm, w_idx, w_cnt);
  k_scatter<<<dim3(CC, BB), 256, 0, stream>>>(features, w_idx, w_cnt, out_vox);
}